// GAT_76716705841462
// MI455X (gfx1250) — compile-verified
//
#include <hip/hip_runtime.h>
#include <math.h>

typedef __attribute__((ext_vector_type(2))) float v2f;
typedef __attribute__((ext_vector_type(8))) float v8f;

#define NNODE 50000
#define NEDGE 300000
#define DIN   256
#define DOUT  256          // H*C
#define NHEAD 2
#define CH    128
#define NEG_SLOPE 0.2f

// ---------------------------------------------------------------------------
// Kernel 1: dual GEMM  h_src = x@W_src, h_dst = x@W_dst  via v_wmma_f32_16x16x4_f32
// grid = 3125 blocks (one 16-row strip of x each), block = 256 threads = 8 waves.
// Waves 0..3 -> W_src column tiles, waves 4..7 -> W_dst column tiles (4 tiles each).
// ---------------------------------------------------------------------------
__global__ __launch_bounds__(256) void gemm_dual_wmma(
    const float* __restrict__ x,
    const float* __restrict__ Wsrc,
    const float* __restrict__ Wdst,
    float* __restrict__ hsrc,
    float* __restrict__ hdst)
{
    __shared__ float xs[16 * DIN];          // 16 KB tile of x

    const int m0 = blockIdx.x * 16;

    // cooperative stage of 16x256 x-tile into LDS (4096 floats = 1024 float4)
    {
        const float4* __restrict__ xg = (const float4*)(x + (size_t)m0 * DIN);
        float4* xsv = (float4*)xs;
        #pragma unroll
        for (int i = 0; i < 4; ++i)
            xsv[threadIdx.x + i * 256] = xg[threadIdx.x + i * 256];
    }
    __syncthreads();

    const int wave = threadIdx.x >> 5;
    const int lane = threadIdx.x & 31;

    const float* __restrict__ W    = (wave < 4) ? Wsrc : Wdst;
    float*       __restrict__ Hout = (wave < 4) ? hsrc : hdst;
    const int ntBase  = (wave & 3) * 4;     // first of 4 column tiles
    const int colLane = lane & 15;          // N within tile
    const int row     = lane & 15;          // M within tile (A frag)
    const int kk      = (lane >> 4) * 2;    // K sub-offset per ISA A/B layout

    v8f acc[4];
    #pragma unroll
    for (int nt = 0; nt < 4; ++nt) acc[nt] = (v8f){};

    for (int k = 0; k < DIN; k += 4) {
        // A fragment: lanes 0-15 hold x[m0+row][k..k+1], lanes 16-31 hold k+2..k+3
        v2f a = *(const v2f*)(&xs[row * DIN + k + kk]);
        #pragma unroll
        for (int nt = 0; nt < 4; ++nt) {
            const int col = (ntBase + nt) * 16 + colLane;
            v2f b;
            b.x = W[(size_t)(k + kk)     * DOUT + col];
            b.y = W[(size_t)(k + kk + 1) * DOUT + col];
            acc[nt] = __builtin_amdgcn_wmma_f32_16x16x4_f32(
                false, a, false, b, (short)0, acc[nt], false, false);
        }
    }

    // store D: VGPR r -> row m0 + (lane>=16?8:0) + r, col = tile*16 + lane%16
    const int rowBase = m0 + (lane >> 4) * 8;
    #pragma unroll
    for (int nt = 0; nt < 4; ++nt) {
        const int col = (ntBase + nt) * 16 + colLane;
        #pragma unroll
        for (int r = 0; r < 8; ++r)
            Hout[(size_t)(rowBase + r) * DOUT + col] = acc[nt][r];
    }
}

// ---------------------------------------------------------------------------
// Kernel 2: per-node attention projections a_src[n][h], a_dst[n][h]
// one wave per (node, head); lanes reduce 128 channels.
// ---------------------------------------------------------------------------
__global__ __launch_bounds__(256) void node_att(
    const float* __restrict__ hsrc, const float* __restrict__ hdst,
    const float* __restrict__ att_src, const float* __restrict__ att_dst,
    float* __restrict__ a_src, float* __restrict__ a_dst)
{
    const int gw   = (blockIdx.x * blockDim.x + threadIdx.x) >> 5;  // (n,h) id
    const int lane = threadIdx.x & 31;
    if (gw >= NNODE * NHEAD) return;
    const int n = gw >> 1, h = gw & 1;

    const float* hr = hsrc + (size_t)n * DOUT + h * CH;
    const float* hd = hdst + (size_t)n * DOUT + h * CH;
    const float* as = att_src + h * CH;
    const float* ad = att_dst + h * CH;

    float s1 = 0.f, s2 = 0.f;
    #pragma unroll
    for (int c = 0; c < CH; c += 32) {
        s1 += hr[c + lane] * as[c + lane];
        s2 += hd[c + lane] * ad[c + lane];
    }
    #pragma unroll
    for (int off = 16; off > 0; off >>= 1) {
        s1 += __shfl_down(s1, off, 32);
        s2 += __shfl_down(s2, off, 32);
    }
    if (lane == 0) { a_src[gw] = s1; a_dst[gw] = s2; }
}

// ---------------------------------------------------------------------------
// Init kernels
// ---------------------------------------------------------------------------
__global__ __launch_bounds__(256) void init_out(float* __restrict__ out,
                                                const float* __restrict__ bias)
{
    const int i = blockIdx.x * 256 + threadIdx.x;     // exactly N*256 threads
    out[i] = 5.0f * bias[i & (DOUT - 1)];             // bias added once per relation
}

__global__ __launch_bounds__(256) void init_ms(float* __restrict__ m,
                                               float* __restrict__ s)
{
    const int i = blockIdx.x * 256 + threadIdx.x;
    if (i < NNODE * NHEAD) { m[i] = -INFINITY; s[i] = 0.f; }
}

// ---------------------------------------------------------------------------
// Segment softmax passes (per relation, self-loops appended as edges E..E+N-1)
// ---------------------------------------------------------------------------
__device__ __forceinline__ void atomicMaxF(float* addr, float val)
{
    if (val >= 0.0f) atomicMax((int*)addr, __float_as_int(val));
    else             atomicMin((unsigned int*)addr, (unsigned int)__float_as_int(val));
}

__device__ __forceinline__ void edge_nodes(const int* __restrict__ ei, int e,
                                           int& src, int& dst)
{
    if (e < NEDGE) { src = ei[e]; dst = ei[NEDGE + e]; }
    else           { src = dst = e - NEDGE; }
}

__device__ __forceinline__ float leaky(float v)
{
    return v >= 0.f ? v : NEG_SLOPE * v;
}

__global__ __launch_bounds__(256) void edge_max(
    const int* __restrict__ ei,
    const float* __restrict__ a_src, const float* __restrict__ a_dst,
    float* __restrict__ m)
{
    const int t = blockIdx.x * 256 + threadIdx.x;
    if (t >= (NEDGE + NNODE) * NHEAD) return;
    const int e = t >> 1, h = t & 1;
    int src, dst; edge_nodes(ei, e, src, dst);
    const float l = leaky(a_src[src * NHEAD + h] + a_dst[dst * NHEAD + h]);
    atomicMaxF(&m[dst * NHEAD + h], l);
}

__global__ __launch_bounds__(256) void edge_expsum(
    const int* __restrict__ ei,
    const float* __restrict__ a_src, const float* __restrict__ a_dst,
    const float* __restrict__ m, float* __restrict__ s)
{
    const int t = blockIdx.x * 256 + threadIdx.x;
    if (t >= (NEDGE + NNODE) * NHEAD) return;
    const int e = t >> 1, h = t & 1;
    int src, dst; edge_nodes(ei, e, src, dst);
    const float l  = leaky(a_src[src * NHEAD + h] + a_dst[dst * NHEAD + h]);
    const float ev = __expf(l - m[dst * NHEAD + h]);
    atomicAdd(&s[dst * NHEAD + h], ev);
}

// 64 threads per edge, 4 channels each; head is uniform within a wave.
__global__ __launch_bounds__(256) void edge_scatter(
    const int* __restrict__ ei,
    const float* __restrict__ a_src, const float* __restrict__ a_dst,
    const float* __restrict__ m, const float* __restrict__ s,
    const float* __restrict__ hsrc, float* __restrict__ out)
{
    const int t = blockIdx.x * 256 + threadIdx.x;     // exactly (E+N)*64 threads
    const int e = t >> 6;
    const int j = (t & 63) * 4;                       // channel group 0..252
    const int h = j >> 7;                             // wave-uniform head
    int src, dst; edge_nodes(ei, e, src, dst);

    const float l     = leaky(a_src[src * NHEAD + h] + a_dst[dst * NHEAD + h]);
    const float alpha = __expf(l - m[dst * NHEAD + h]) /
                        (s[dst * NHEAD + h] + 1e-16f);

    const float4 hv = *(const float4*)(hsrc + (size_t)src * DOUT + j);
    float* o = out + (size_t)dst * DOUT + j;
    atomicAdd(o + 0, alpha * hv.x);
    atomicAdd(o + 1, alpha * hv.y);
    atomicAdd(o + 2, alpha * hv.z);
    atomicAdd(o + 3, alpha * hv.w);
}

// ---------------------------------------------------------------------------
extern "C" void kernel_launch(void* const* d_in, const int* in_sizes, int n_in,
                              void* d_out, int out_size, void* d_ws, size_t ws_size,
                              hipStream_t stream)
{
    const float* x       = (const float*)d_in[0];
    const int*   edges[5] = { (const int*)d_in[1], (const int*)d_in[2],
                              (const int*)d_in[3], (const int*)d_in[4],
                              (const int*)d_in[5] };
    const float* Wsrc    = (const float*)d_in[6];
    const float* Wdst    = (const float*)d_in[7];
    const float* att_src = (const float*)d_in[8];
    const float* att_dst = (const float*)d_in[9];
    const float* bias    = (const float*)d_in[10];
    float* out = (float*)d_out;

    // workspace layout (floats)
    float* ws     = (float*)d_ws;
    float* hsrc   = ws;                              // N*256
    float* hdst   = hsrc + (size_t)NNODE * DOUT;     // N*256
    float* a_src  = hdst + (size_t)NNODE * DOUT;     // N*H
    float* a_dst  = a_src + NNODE * NHEAD;
    float* mbuf   = a_dst + NNODE * NHEAD;
    float* sbuf   = mbuf  + NNODE * NHEAD;

    // 1) linear transforms via WMMA
    gemm_dual_wmma<<<NNODE / 16, 256, 0, stream>>>(x, Wsrc, Wdst, hsrc, hdst);

    // 2) per-node attention coefficients
    node_att<<<(NNODE * NHEAD) / 8, 256, 0, stream>>>(hsrc, hdst, att_src, att_dst,
                                                      a_src, a_dst);

    // 3) out = 5*bias (each relation contributes bias once)
    init_out<<<(NNODE * DOUT) / 256, 256, 0, stream>>>(out, bias);

    // 4) five relations: segment softmax + weighted scatter
    const int nEH  = (NEDGE + NNODE) * NHEAD;            // 700000
    const int nSC  = (NEDGE + NNODE) * 64;               // 22.4M
    for (int r = 0; r < 5; ++r) {
        init_ms<<<(NNODE * NHEAD + 255) / 256, 256, 0, stream>>>(mbuf, sbuf);
        edge_max    <<<(nEH + 255) / 256, 256, 0, stream>>>(edges[r], a_src, a_dst, mbuf);
        edge_expsum <<<(nEH + 255) / 256, 256, 0, stream>>>(edges[r], a_src, a_dst, mbuf, sbuf);
        edge_scatter<<<nSC / 256, 256, 0, stream>>>(edges[r], a_src, a_dst,
                                                    mbuf, sbuf, hsrc, out);
    }
}